// Capsule_80272938762493
// MI455X (gfx1250) — compile-verified
//
#include <hip/hip_runtime.h>

typedef __attribute__((ext_vector_type(2))) float v2f;
typedef __attribute__((ext_vector_type(8))) float v8f;

#define NCAP 5
#define DCAP 5
#define OCAP 25      // NCAP*DCAP
#define FDIM 80
#define IDIM 256
#define NROUTE 4
#define CAPS_EPS 1e-7f

#define W_STRIDE 84  // padded LDS row stride for W (dwords): conflict-free for 16 lanes
#define U_STRIDE 29  // padded LDS row stride for u_hat: conflict-free for WMMA scatter + row reads

__global__ __launch_bounds__(256) void capsule_routing_kernel(
    const float* __restrict__ x,   // [B, 256, 80]
    const float* __restrict__ W,   // [25, 80]
    float* __restrict__ out) {     // [B, 5, 5]
  __shared__ float Wlds[32 * W_STRIDE];          // 10.5 KB, rows 25..31 zero
  __shared__ float Ulds[IDIM * U_STRIDE];        // 29 KB u_hat tile for this b
  __shared__ float sred[OCAP];                   // routing reduction target

  const int tid = threadIdx.x;
  const int b   = blockIdx.x;
  const float* xb = x + (size_t)b * IDIM * FDIM;

  // ---- stage W into LDS; zero-pad rows 25..31 so N-tile 1 needs no masking ----
  for (int idx = tid; idx < OCAP * FDIM; idx += 256) {
    int o = idx / FDIM, f = idx - o * FDIM;
    Wlds[o * W_STRIDE + f] = W[idx];
  }
  for (int idx = tid; idx < (32 - OCAP) * W_STRIDE; idx += 256) {
    Wlds[OCAP * W_STRIDE + idx] = 0.0f;
  }
  __syncthreads();

  // ---- phase 1: u_hat = x[b] (256x80) @ W^T (80x25) via V_WMMA_F32_16X16X4_F32 ----
  const int wave = tid >> 5;          // 8 waves
  const int lane = tid & 31;
  const int m    = lane & 15;         // A row within tile / B-C column
  const int kh   = (lane >> 4) << 1;  // lanes 0-15 hold K={0,1}, lanes 16-31 K={2,3}
  const int hi8  = (lane >> 4) << 3;  // C/D: hi half lanes hold rows M=8..15

  for (int t = 0; t < 2; ++t) {
    const int R = wave * 32 + t * 16;                    // 16-row tile base (covers 256 rows)
    const float* arow = xb + (R + m) * FDIM + kh;        // A fragment source (global, read-once)
    const float* b0p  = &Wlds[m * W_STRIDE + kh];        // B fragment, N columns 0..15
    const float* b1p  = &Wlds[(16 + m) * W_STRIDE + kh]; // B fragment, N columns 16..31 (zero pad)
    v8f acc0 = {};
    v8f acc1 = {};
#pragma unroll
    for (int kb = 0; kb < FDIM / 4; ++kb) {              // 20 K-blocks of 4
      v2f a  = *(const v2f*)(arow + kb * 4);
      v2f b0 = *(const v2f*)(b0p  + kb * 4);
      v2f b1 = *(const v2f*)(b1p  + kb * 4);
      acc0 = __builtin_amdgcn_wmma_f32_16x16x4_f32(false, a, false, b0, (short)0, acc0, false, false);
      acc1 = __builtin_amdgcn_wmma_f32_16x16x4_f32(false, a, false, b1, (short)0, acc1, false, false);
    }
    // scatter accumulators: VGPR r holds row M = r + hi8, column n = m (+16 for tile 1)
#pragma unroll
    for (int r = 0; r < 8; ++r) {
      Ulds[(R + r + hi8) * U_STRIDE + m] = acc0[r];
      if (16 + m < OCAP) Ulds[(R + r + hi8) * U_STRIDE + 16 + m] = acc1[r];
    }
  }
  __syncthreads();

  // ---- phase 2: dynamic routing, thread i owns input capsule i ----
  float u[OCAP];
#pragma unroll
  for (int o = 0; o < OCAP; ++o) u[o] = Ulds[tid * U_STRIDE + o];

  float blog[NCAP];
#pragma unroll
  for (int n = 0; n < NCAP; ++n) blog[n] = 0.0f;

  float v[OCAP];

  for (int it = 0; it < NROUTE; ++it) {
    // softmax over the 5 output capsules (per input capsule i)
    float mx = blog[0];
#pragma unroll
    for (int n = 1; n < NCAP; ++n) mx = fmaxf(mx, blog[n]);
    float c[NCAP];
    float esum = 0.0f;
#pragma unroll
    for (int n = 0; n < NCAP; ++n) { c[n] = __expf(blog[n] - mx); esum += c[n]; }
    float inv = 1.0f / esum;
#pragma unroll
    for (int n = 0; n < NCAP; ++n) c[n] *= inv;

    // s[n,d] = sum_i c[n,i] * u_hat[n,i,d] : shuffle tree + LDS float atomics
    if (tid < OCAP) sred[tid] = 0.0f;
    __syncthreads();
#pragma unroll
    for (int o = 0; o < OCAP; ++o) {
      float p = c[o / DCAP] * u[o];
      p += __shfl_xor(p, 16);
      p += __shfl_xor(p, 8);
      p += __shfl_xor(p, 4);
      p += __shfl_xor(p, 2);
      p += __shfl_xor(p, 1);
      if (lane == 0) atomicAdd(&sred[o], p);   // ds_add_f32
    }
    __syncthreads();

    // squash (computed redundantly by every thread from the 25-float LDS vector)
#pragma unroll
    for (int n = 0; n < NCAP; ++n) {
      float sn = 0.0f;
#pragma unroll
      for (int d = 0; d < DCAP; ++d) { float sv = sred[n * DCAP + d]; sn += sv * sv; }
      float scale = sn / ((1.0f + sn) * sqrtf(sn + CAPS_EPS));
#pragma unroll
      for (int d = 0; d < DCAP; ++d) v[n * DCAP + d] = scale * sred[n * DCAP + d];
    }

    if (it < NROUTE - 1) {
      // b[n,i] += sum_d outputs[n,d] * u_hat[n,i,d]
#pragma unroll
      for (int n = 0; n < NCAP; ++n) {
        float acc = 0.0f;
#pragma unroll
        for (int d = 0; d < DCAP; ++d) acc = fmaf(v[n * DCAP + d], u[n * DCAP + d], acc);
        blog[n] += acc;
      }
      __syncthreads();  // protect sred before next iteration's re-zero
    }
  }

  if (tid < OCAP) out[(size_t)b * OCAP + tid] = v[tid];
}

extern "C" void kernel_launch(void* const* d_in, const int* in_sizes, int n_in,
                              void* d_out, int out_size, void* d_ws, size_t ws_size,
                              hipStream_t stream) {
  const float* x = (const float*)d_in[0];
  const float* W = (const float*)d_in[1];
  float* out = (float*)d_out;
  const int B = in_sizes[0] / (IDIM * FDIM);   // 8192
  capsule_routing_kernel<<<dim3(B), dim3(256), 0, stream>>>(x, W, out);
}